// ReaReaConv_28492813041739
// MI455X (gfx1250) — compile-verified
//
#include <hip/hip_runtime.h>
#include <hip/hip_bf16.h>

#define N_NODES 50000
#define N_EDGES 1600000
#define BATCH   2
#define CCH     64   // C_IN == C_OUT == 64

typedef float v2f __attribute__((ext_vector_type(2)));
typedef float v8f __attribute__((ext_vector_type(8)));

// ---------------------------------------------------------------------------
// 1) deg[i] = 1 (self loop)
__global__ void rrc_init_deg(float* __restrict__ deg) {
    int i = blockIdx.x * blockDim.x + threadIdx.x;
    if (i < N_NODES) deg[i] = 1.0f;
}

// 2) deg[tgt[e]] += 1
__global__ void rrc_count_deg(const int* __restrict__ tgt, float* __restrict__ deg) {
    int e = blockIdx.x * blockDim.x + threadIdx.x;
    if (e < N_EDGES) atomicAdd(&deg[tgt[e]], 1.0f);
}

// 3) dis[i] = rsqrt(deg[i])   (deg >= 1 always due to self loops)
__global__ void rrc_finalize_dis(float* __restrict__ deg) {
    int i = blockIdx.x * blockDim.x + threadIdx.x;
    if (i < N_NODES) {
        float d = deg[i];
        deg[i] = (d > 0.0f) ? rsqrtf(d) : 0.0f;
    }
}

// ---------------------------------------------------------------------------
// 4) WMMA f32 GEMM:  y = x * W^T   (one weight matrix per gridDim.y slice)
//    One wave32 handles TILES_PER_WAVE 16-row tiles. The single weight
//    matrix fragments into 64 v2f = 128 VGPRs/lane and stays register-
//    resident (total footprint ~210 VGPRs < 256, so no scratch spills).
//    Per tile: 16 clause-batched A loads, then 64 back-to-back WMMAs.
#define TILES_PER_WAVE 8

__global__ __launch_bounds__(128, 1)
void rrc_gemm_wmma(const float* __restrict__ x,
                   const float* __restrict__ Wc,
                   const float* __restrict__ Wd,
                   float* __restrict__ xc,
                   float* __restrict__ xd) {
    const int ROWS  = BATCH * N_NODES;            // 100000, multiple of 16
    const int TILES = ROWS / 16;                  // 6250

    int wave  = (blockIdx.x * blockDim.x + threadIdx.x) >> 5;
    int tile0 = wave * TILES_PER_WAVE;
    if (tile0 >= TILES) return;                   // wave-uniform; EXEC stays all-1s

    // gridDim.y = 2: slice 0 -> (Wc, xc), slice 1 -> (Wd, xd). Block-uniform.
    const float* __restrict__ W = (blockIdx.y == 0) ? Wc : Wd;
    float*       __restrict__ y = (blockIdx.y == 0) ? xc : xd;

    int lane = threadIdx.x & 31;
    int lo   = lane & 15;                         // M (A) / N (B,C,D) index
    int hi   = lane >> 4;                         // K-half (A,B) / M-half (C,D)
    int kl   = hi * 2;                            // per-lane K base within a K=4 chunk

    // ---- Preload all B fragments of this matrix (W row-major: W[n][k]) ----
    v2f Bf[16 * 4];                               // 128 VGPRs, register-resident
#pragma unroll
    for (int kk = 0; kk < 16; ++kk) {
#pragma unroll
        for (int t = 0; t < 4; ++t) {
            Bf[kk * 4 + t] = *(const v2f*)(W + (size_t)(t * 16 + lo) * CCH + kk * 4 + kl);
        }
    }

    for (int ti = 0; ti < TILES_PER_WAVE; ++ti) {
        int tile = tile0 + ti;
        if (tile >= TILES) break;                 // wave-uniform
        int rowBase = tile * 16;
        const float* xrow = x + (size_t)(rowBase + lo) * CCH;

        // Prefetch next tile's A row (speculative; OOB silently dropped)
        __builtin_prefetch(xrow + 16 * CCH, 0, 0);

        // Hoist all 16 A-fragment loads (one clause, one wait)
        v2f a[16];
#pragma unroll
        for (int kk = 0; kk < 16; ++kk)
            a[kk] = *(const v2f*)(xrow + kk * 4 + kl);

        v8f acc[4] = {};
#pragma unroll
        for (int kk = 0; kk < 16; ++kk) {
#pragma unroll
            for (int t = 0; t < 4; ++t) {         // reuse distance 4 on each acc
                acc[t] = __builtin_amdgcn_wmma_f32_16x16x4_f32(
                    false, a[kk], false, Bf[kk * 4 + t], (short)0, acc[t], false, false);
            }
        }

#pragma unroll
        for (int t = 0; t < 4; ++t) {
#pragma unroll
            for (int r = 0; r < 8; ++r) {
                int m = rowBase + r + hi * 8;     // C/D layout: VGPR r -> M = r + 8*hi
                y[(size_t)m * CCH + t * 16 + lo] = acc[t][r];
            }
        }
    }
}

// ---------------------------------------------------------------------------
// 5) out = bias + (self-loop term) = bias[c] + xc[row][c] * dis[n]^2
__global__ void rrc_init_out(const float* __restrict__ dis,
                             const float* __restrict__ bias,
                             const float* __restrict__ xc,
                             float* __restrict__ out) {
    int i = blockIdx.x * blockDim.x + threadIdx.x;      // over B*N*C = 6,400,000
    if (i >= BATCH * N_NODES * CCH) return;
    int c   = i & (CCH - 1);
    int row = i >> 6;                                   // b*N + n
    int n   = row % N_NODES;
    float d = dis[n];
    out[i] = bias[c] + xc[i] * d * d;
}

// ---------------------------------------------------------------------------
// 6) Edge scatter: one wave32 per edge (blockIdx.y = batch), 2 channels/lane.
__global__ void rrc_scatter(const int*   __restrict__ src,
                            const int*   __restrict__ tgt,
                            const float* __restrict__ f_disc_orig,
                            const float* __restrict__ fluxes,
                            const float* __restrict__ dis,
                            const float* __restrict__ xc,
                            const float* __restrict__ xd,
                            float*       __restrict__ out) {
    long long gtid = (long long)blockIdx.x * blockDim.x + threadIdx.x;
    long long e    = gtid >> 5;
    if (e >= N_EDGES) return;
    int lane = (int)(gtid & 31);
    int b    = blockIdx.y;

    int s = src[e];
    int t = tgt[e];

    float fs   = fluxes[(size_t)b * N_NODES + s];
    float ft   = fluxes[(size_t)b * N_NODES + t];
    float keep = 0.5f * (1.0f + tanhf(fs * ft));        // TEMPERATURE = 1.0
    float fdo  = f_disc_orig[e];
    float f    = keep * fdo + (1.0f - keep) * (1.0f - fdo);
    float norm = dis[s] * dis[t];
    float wc   = norm * (1.0f - f);
    float wd   = norm * f;

    size_t srow = ((size_t)b * N_NODES + s) * CCH;
    size_t trow = ((size_t)b * N_NODES + t) * CCH;
    int c0 = lane * 2;

    v2f vc = *(const v2f*)(xc + srow + c0);
    v2f vd = *(const v2f*)(xd + srow + c0);

    atomicAdd(&out[trow + c0    ], wc * vc.x + wd * vd.x);
    atomicAdd(&out[trow + c0 + 1], wc * vc.y + wd * vd.y);
}

// ---------------------------------------------------------------------------
extern "C" void kernel_launch(void* const* d_in, const int* in_sizes, int n_in,
                              void* d_out, int out_size, void* d_ws, size_t ws_size,
                              hipStream_t stream) {
    const float* x           = (const float*)d_in[0];   // (B, N, 64)
    const int*   edge_index  = (const int*)  d_in[1];   // (2, E)
    const float* f_disc_orig = (const float*)d_in[2];   // (E,)
    const float* fluxes      = (const float*)d_in[3];   // (B, N)
    const float* Wc          = (const float*)d_in[4];   // (64, 64)
    const float* Wd          = (const float*)d_in[5];   // (64, 64)
    const float* bias        = (const float*)d_in[6];   // (64,)
    float*       out         = (float*)d_out;           // (B, N, 64)

    const int* src = edge_index;
    const int* tgt = edge_index + N_EDGES;

    // Workspace: dis (N f32) | xc (B*N*64 f32) | xd (B*N*64 f32)  ~= 52 MB
    float* dis = (float*)d_ws;                           // also holds deg first
    float* xc  = dis + N_NODES;                          // 200000 B offset (16B aligned)
    float* xd  = xc + (size_t)BATCH * N_NODES * CCH;

    // 1-3: degree -> dis
    rrc_init_deg    <<<(N_NODES + 255) / 256, 256, 0, stream>>>(dis);
    rrc_count_deg   <<<(N_EDGES + 255) / 256, 256, 0, stream>>>(tgt, dis);
    rrc_finalize_dis<<<(N_NODES + 255) / 256, 256, 0, stream>>>(dis);

    // 4: WMMA GEMM (6250 tiles, 8 tiles/wave, 4 waves/block; grid.y selects W)
    {
        const int TILES = (BATCH * N_NODES) / 16;        // 6250
        const int WAVES = (TILES + TILES_PER_WAVE - 1) / TILES_PER_WAVE;  // 782
        const int WPB   = 4;
        dim3 grid((WAVES + WPB - 1) / WPB, 2, 1);
        rrc_gemm_wmma<<<grid, WPB * 32, 0, stream>>>(x, Wc, Wd, xc, xd);
    }

    // 5: out = bias + self-loop contribution
    {
        const int TOT = BATCH * N_NODES * CCH;           // 6.4M
        rrc_init_out<<<(TOT + 255) / 256, 256, 0, stream>>>(dis, bias, xc, out);
    }

    // 6: edge scatter (1 wave per edge, grid.y = batch)
    {
        long long threads = (long long)N_EDGES * 32;     // 51.2M per batch
        dim3 grid((unsigned)((threads + 255) / 256), BATCH, 1);
        rrc_scatter<<<grid, 256, 0, stream>>>(src, tgt, f_disc_orig, fluxes,
                                              dis, xc, xd, out);
    }
}